// blendv1_39453569581121
// MI455X (gfx1250) — compile-verified
//
#include <hip/hip_runtime.h>

// Fused HMETS/HBV/VIC hydrology scan + gamma-UH routing for MI455X (gfx1250).
// One thread per grid cell; all state, parameters, UH weights and the 15-deep
// routing accumulator live in VGPRs (wave32; only 128 waves exist, so register
// pressure is free). Single pass over x, single write of flow. The x stream is
// software-pipelined one step ahead (register double-buffer) plus a 16-step
// global_prefetch_b8 lookahead so the serial state chain never waits on memory.

namespace {
constexpr int T_STEPS = 512;
constexpr int G_CELLS = 4096;
constexpr int LUH     = 15;
constexpr int NPAR    = 39;
constexpr int BLOCK   = 128;
constexpr float NZf   = 1e-5f;
}

// Native gfx1250 transcendentals: v_exp_f32 / v_log_f32 (base-2).
__device__ __forceinline__ float fexp2(float x) { return __builtin_amdgcn_exp2f(x); }
__device__ __forceinline__ float flog2(float x) { return __builtin_amdgcn_logf(x); }
__device__ __forceinline__ float fexpe(float x) { return fexp2(x * 1.4426950408889634f); } // e^x
__device__ __forceinline__ float sigf(float z)  { return 1.0f / (1.0f + fexpe(-z)); }
__device__ __forceinline__ float p10f(float z)  { return fexp2(z * 3.3219280948873623f); }  // 10^z
__device__ __forceinline__ float powpf(float a, float b) { return fexp2(b * flog2(a)); }    // a > 0 guaranteed

// CDNA5 async global->LDS path (ASYNCcnt-tracked), guarded so the compile is
// safe on toolchains that do not expose the builtins. Probe confirmed this
// toolchain declares it with generic (flat) int* parameters.
#if defined(__has_builtin)
#if __has_builtin(__builtin_amdgcn_global_load_async_to_lds_b32)
#define USE_ASYNC_LDS 1
#endif
#endif

__global__ void __launch_bounds__(BLOCK, 1)
hydro_fused_kernel(const float* __restrict__ x,      // (T, G, 3)
                   const float* __restrict__ params, // (T, G, 39) -- only t = T-1 used
                   float* __restrict__ flow)         // (T, G)
{
    __shared__ float sprm[BLOCK * NPAR];
    const int g0 = blockIdx.x * BLOCK;
    const int g  = g0 + threadIdx.x;

    // Coalesced stage of the last-timestep parameter rows through LDS
    // (per-thread stride-39 reads from global would be fully uncoalesced).
    {
        const float* pb = params + ((size_t)(T_STEPS - 1) * G_CELLS + (size_t)g0) * NPAR;
#ifdef USE_ASYNC_LDS
        // dword-granular async copies: the 39-float row stride keeps everything
        // 4B-aligned (16B chunks would be misaligned). 39 iterations, all lanes on.
        for (int i = threadIdx.x; i < BLOCK * NPAR; i += BLOCK) {
            __builtin_amdgcn_global_load_async_to_lds_b32((int*)(pb + i), (int*)(sprm + i), 0, 0);
        }
#if __has_builtin(__builtin_amdgcn_s_wait_asynccnt)
        __builtin_amdgcn_s_wait_asynccnt(0);
#else
        asm volatile("s_wait_asynccnt 0x0" ::: "memory");
#endif
#else
        for (int i = threadIdx.x; i < BLOCK * NPAR; i += BLOCK) sprm[i] = pb[i];
#endif
    }
    __syncthreads();
    float P[NPAR];
#pragma unroll
    for (int i = 0; i < NPAR; ++i) P[i] = sprm[threadIdx.x * NPAR + i]; // gcd(39,64)=1 -> bank-conflict-free

    // ---- descale physical parameters (sigmoid -> bounds) ----
    const float ddf_min  = 20.0f   * sigf(P[0]);
    const float ddf_plus = 20.0f   * sigf(P[1]);
    const float Kcum     = 0.01f  + 0.19f   * sigf(P[2]);
    const float Kf       = 5.0f    * sigf(P[3]);
    const float exp_fe   =           sigf(P[4]);
    const float Tbf      = -5.0f  + 7.0f    * sigf(P[5]);
    const float Ccum     = 0.005f + 0.045f  * sigf(P[6]);
    const float SWI      = 0.4f    * sigf(P[7]);
    const float Tbm      = -2.0f  + 5.0f    * sigf(P[8]);
    const float fcmin    = 0.1f    * sigf(P[9]);
    const float fcminp   = 0.01f  + 0.24f   * sigf(P[10]);
    const float vmax     = 0.001f + 499.999f* sigf(P[11]);
    const float halpha   =           sigf(P[12]);
    const float vic_beta = 0.1f   + 2.9f    * sigf(P[13]);
    const float hbv_beta = 0.5f   + 2.5f    * sigf(P[14]);
    const float qk       = -5.0f  + 3.0f    * sigf(P[15]);
    const float qn       = 0.5f   + 1.5f    * sigf(P[16]);
    const float mmax     = 100.0f  * sigf(P[17]);
    const float lamb     = 5.0f   + 5.0f    * sigf(P[18]);
    const float bk       = -5.0f  + 4.0f    * sigf(P[19]);
    const float bn       = 0.5f   + 1.5f    * sigf(P[20]);
    const float ET_eff   = 3.0f    * sigf(P[21]);
    const float c_v2p    = 1e-5f  + (0.02f - 1e-5f) * sigf(P[22]);
    // P[23] (coef_phreatic) is unpacked by the reference but never used in step().

    // ---- softmax weight groups ----
    float w1, w2, w3, w4, w5, w6, w7, w8, w9, w10, w11;
    {
        float a = P[24], b = P[25], c = P[26];
        float m = fmaxf(a, fmaxf(b, c));
        float ea = fexpe(a - m), eb = fexpe(b - m), ec = fexpe(c - m);
        float s = 1.0f / (ea + eb + ec); w1 = ea * s; w2 = eb * s; w3 = ec * s;
    }
    {
        float a = P[27], b = P[28], c = P[29];
        float m = fmaxf(a, fmaxf(b, c));
        float ea = fexpe(a - m), eb = fexpe(b - m), ec = fexpe(c - m);
        float s = 1.0f / (ea + eb + ec); w4 = ea * s; w5 = eb * s; w6 = ec * s;
    }
    {
        float a = P[30], b = P[31], c = P[32];
        float m = fmaxf(a, fmaxf(b, c));
        float ea = fexpe(a - m), eb = fexpe(b - m), ec = fexpe(c - m);
        float s = 1.0f / (ea + eb + ec); w7 = ea * s; w8 = eb * s; w9 = ec * s;
    }
    {
        float a = P[33], b = P[34];
        float m = fmaxf(a, b);
        float ea = fexpe(a - m), eb = fexpe(b - m);
        float s = 1.0f / (ea + eb); w10 = ea * s; w11 = eb * s;
    }

    // ---- gamma unit hydrographs (per cell, 15 taps each) ----
    const float a1 = 0.3f  + 19.7f * sigf(P[35]);
    const float b1 = 0.01f + 4.99f * sigf(P[36]);
    const float a2 = 0.5f  + 12.5f * sigf(P[37]);
    const float b2 = 0.15f + 1.35f * sigf(P[38]);
    float uh1[LUH], uh2[LUH];
    {
        const float LOG2E = 1.4426950408889634f;
        float s1 = 0.0f, s2 = 0.0f;
#pragma unroll
        for (int k = 0; k < LUH; ++k) {
            float tk  = (float)k + 0.5f;
            float l2t = flog2(tk);
            float e1  = fexp2((a1 - 1.0f) * l2t - (tk / b1) * LOG2E); // t^(a-1) e^{-t/b}
            float e2  = fexp2((a2 - 1.0f) * l2t - (tk / b2) * LOG2E);
            uh1[k] = e1; uh2[k] = e2; s1 += e1; s2 += e2;
        }
        float r1 = 1.0f / s1, r2 = 1.0f / s2;
#pragma unroll
        for (int k = 0; k < LUH; ++k) { uh1[k] *= r1; uh2[k] *= r2; }
    }

    // ---- derived loop-invariant constants ----
    const float ddf_cap  = ddf_min + ddf_plus;
    const float p10qk    = p10f(qk);
    const float p10bk    = p10f(bk);
    const float ctop     = mmax * (vmax / qn) * fexp2(-qn * flog2(lamb)); // mmax*(vmax/qn)/lamb^qn
    const float inv_vmax = 1.0f / vmax;

    // ---- state (all VGPR-resident) ----
    float sp1 = 0, lw1 = 0, cm1 = 0;
    float sp2 = 0, lw2 = 0, cm2 = 0;
    float sp3 = 0, cm3 = 0;
    float vad = 0, phr = 0;
    float acc[LUH];
#pragma unroll
    for (int k = 0; k < LUH; ++k) acc[k] = 0.0f;

    const float* xg = x + (size_t)g * 3;

    // Software pipeline: hold step-t inputs in registers, issue step-(t+1)
    // loads before the long dependent chain so s_wait_loadcnt sinks to the end.
    float cur0 = xg[0], cur1 = xg[1], cur2 = xg[2];

#pragma unroll 1
    for (int t = 0; t < T_STEPS; ++t) {
        const int nt = (t + 1 < T_STEPS) ? (t + 1) : t;
        const int nbase = nt * (G_CELLS * 3);
        float nx0 = xg[nbase + 0];
        float nx1 = xg[nbase + 1];
        float nx2 = xg[nbase + 2];
        {   // stream x ahead of the dependent chain: gfx1250 global_prefetch_b8
            int tp = (t + 16 < T_STEPS) ? (t + 16) : (T_STEPS - 1);
            __builtin_prefetch(xg + (size_t)tp * (G_CELLS * 3), 0, 0);
        }

        float prcp = cur0;
        float tm   = cur1;
        float pet  = cur2;
        float rain = (tm > 0.0f) ? prcp : 0.0f;
        float snow = prcp - rain;

        // --- snobal_hbv (matches reference exactly, incl. sf+rfz double-count in melt) ---
        float o1;
        {
            float ddf  = fminf(ddf_cap, ddf_min * (1.0f + Kcum * cm1));
            float pm   = fmaxf(ddf * (tm - Tbm), 0.0f);
            float rfz  = fminf(Kf * fmaxf(Tbf - tm, NZf), lw1);
            sp1 = sp1 + snow + rfz;
            float melt = fminf(pm, sp1 + snow + rfz);
            cm1 = (sp1 > NZf) ? (cm1 + melt) : 0.0f;   // uses pre-melt sp
            sp1 = fmaxf(sp1 - melt, NZf);
            float wret = SWI * sp1;
            float wtmp = lw1 + melt + rain;
            float ovf  = fmaxf(wtmp - wret, 0.0f);
            lw1 = (ovf > 0.0f) ? wret : wtmp;
            o1 = ovf;
        }
        // --- snobal_hmets (note: cm updated twice, as in reference) ---
        float o2;
        {
            float ddf  = fminf(ddf_cap, ddf_min * (1.0f + Kcum * cm2));
            float pm   = fmaxf(ddf * (tm - Tbm), 0.0f);
            float pfz  = Kf * powpf(fmaxf(Tbf - tm, NZf), exp_fe);
            float rfz  = fminf(pfz, lw2);
            lw2 = lw2 - rfz;
            sp2 = sp2 + rfz;
            float melt = fminf(pm, sp2 + snow);
            sp2 = sp2 + snow - melt;
            cm2 = (sp2 > NZf) ? (cm2 + melt) : 0.0f;
            float wrf  = fmaxf((fcmin + fcminp) * (1.0f - Ccum * cm2), fcmin);
            float wret = wrf * sp2;
            float wtmp = lw2 + melt + rain;
            float ovf  = fmaxf(wtmp - wret, 0.0f);
            lw2 = (ovf > 0.0f) ? wret : wtmp;
            cm2 = (sp2 > NZf) ? (cm2 + melt) : 0.0f;   // second update per reference
            o2 = ovf;
        }
        // --- snobal_simple ---
        float o3;
        {
            float ddf  = fminf(ddf_cap, ddf_min * (1.0f + Kcum * cm3));
            float pm   = fmaxf(ddf * (tm - Tbm), 0.0f);
            float melt = fminf(pm, sp3);
            sp3 = sp3 + snow - melt;
            o3  = melt + rain;
            cm3 = (sp3 > NZf) ? (cm3 + melt) : 0.0f;
        }

        // --- soil / runoff partition ---
        float rr     = w1 * o1 + w2 * o2 + w3 * o3;
        float sprop  = fminf(fmaxf(vad * inv_vmax, NZf), 1.0f - NZf);
        float l2omsp = flog2(1.0f - sprop);
        float inf = w4 * rr * fexp2(hbv_beta * l2omsp)
                  + w5 * rr * (1.0f - halpha * vad * inv_vmax)
                  + w6 * rr * (1.0f - fexp2(vic_beta * l2omsp));
        inf = fminf(fmaxf(inf, 0.0f), rr);
        float surface = rr - inf;
        vad += inf;
        sprop = fminf(fmaxf(vad * inv_vmax, NZf), 1.0f - NZf);
        float et = fminf(pet * ET_eff * sprop, vad);
        vad -= et;
        float spq    = fexp2(qn * flog2(sprop)); // sprop^qn (shared by top & vic)
        float qf_lin = p10qk * vad;
        float qf_top = fminf(ctop * spq, vad);
        float qf_vic = fminf(mmax * spq, vad);
        float qf     = fminf(w7 * qf_lin + w8 * qf_top + w9 * qf_vic, vad);
        vad -= qf;
        float perc = c_v2p * vad;
        vad -= perc;
        phr += perc;
        float bfv = fminf(p10bk * (w10 * phr + w11 * powpf(fmaxf(phr, NZf), bn)), phr);
        phr -= bfv;

        // --- fused routing: scatter-accumulate into 15 future outputs ---
        float qt = surface + qf;
        float bt = bfv;
#pragma unroll
        for (int j = 0; j < LUH; ++j)
            acc[j] = fmaf(uh1[j], qt, fmaf(uh2[j], bt, acc[j]));
        flow[(size_t)t * G_CELLS + g] = acc[0];
#pragma unroll
        for (int j = 0; j < LUH - 1; ++j) acc[j] = acc[j + 1];
        acc[LUH - 1] = 0.0f;

        cur0 = nx0; cur1 = nx1; cur2 = nx2;   // consume pipelined loads last
    }
}

extern "C" void kernel_launch(void* const* d_in, const int* in_sizes, int n_in,
                              void* d_out, int out_size, void* d_ws, size_t ws_size,
                              hipStream_t stream) {
    (void)in_sizes; (void)n_in; (void)out_size; (void)d_ws; (void)ws_size;
    const float* x      = (const float*)d_in[0]; // (512, 4096, 3)
    const float* params = (const float*)d_in[1]; // (512, 4096, 39)
    float*       flow   = (float*)d_out;         // (512, 4096)
    dim3 grid(G_CELLS / BLOCK), block(BLOCK);
    hipLaunchKernelGGL(hydro_fused_kernel, grid, block, 0, stream, x, params, flow);
}